// SAGE_62130996904578
// MI455X (gfx1250) — compile-verified
//
#include <hip/hip_runtime.h>
#include <math.h>

typedef float v2f __attribute__((ext_vector_type(2)));
typedef float v8f __attribute__((ext_vector_type(8)));

#define EPS 1e-5f

// ---------------- segment-sum scatter (edge aggregation) ----------------
__global__ void scatter_add_kernel(const float* __restrict__ x,
                                   const int* __restrict__ src,
                                   const int* __restrict__ dst,
                                   float* __restrict__ agg,
                                   long long total, int F) {
  long long i = (long long)blockIdx.x * blockDim.x + threadIdx.x;
  if (i >= total) return;
  int e = (int)(i / F);
  int f = (int)(i - (long long)e * F);
  atomicAdd(&agg[(long long)dst[e] * F + f], x[(long long)src[e] * F + f]);
}

__global__ void count_kernel(const int* __restrict__ dst,
                             float* __restrict__ cnt, int E) {
  int i = blockIdx.x * blockDim.x + threadIdx.x;
  if (i < E) atomicAdd(&cnt[dst[i]], 1.0f);
}

__global__ void inv_count_kernel(float* __restrict__ cnt, int n) {
  int i = blockIdx.x * blockDim.x + threadIdx.x;
  if (i < n) {
    float c = cnt[i];
    cnt[i] = 1.0f / (c > 1.0f ? c : 1.0f);
  }
}

// ---------------- dual GEMM: C = (A1 * rowscale) @ B1 + A2 @ B2 + bias ----------------
// One wave32 computes one 16x16 f32 output tile via V_WMMA_F32_16X16X4_F32.
// A1: M x K (segment sums, scaled per-row by inv[]), A2: M x K (x_target, ld == K),
// B1/B2: K x N row-major, bias: N.
__global__ void gemm_dual_wmma(const float* __restrict__ A1,
                               const float* __restrict__ inv,
                               const float* __restrict__ A2,
                               const float* __restrict__ B1,
                               const float* __restrict__ B2,
                               const float* __restrict__ bias,
                               float* __restrict__ C,
                               int M, int N, int K) {
  int wave = (int)((blockIdx.x * (long long)blockDim.x + threadIdx.x) >> 5);
  int lane = threadIdx.x & 31;
  int tilesN = N >> 4;
  int numTiles = (M >> 4) * tilesN;
  if (wave >= numTiles) return;          // wave-uniform: EXEC stays all-1s

  int tm = (wave / tilesN) << 4;
  int tn = (wave % tilesN) << 4;
  int half = lane >> 4;                  // 0: K+0/1,  1: K+2/3
  int l = lane & 15;
  int col = tn + l;                      // C/D and B column for this lane
  int rowA = tm + l;                     // A row for this lane (both halves)

  v8f c;
  {
    float b = bias[col];
#pragma unroll
    for (int r = 0; r < 8; ++r) c[r] = b;
  }

  const float sA = inv[rowA];
  const float* a1p = A1 + (long long)rowA * K;
  const float* a2p = A2 + (long long)rowA * K;

  // pass 1: (A1 * rowscale) @ B1
#pragma unroll 4
  for (int k = 0; k < K; k += 4) {
    int ka = k + (half << 1);
    v2f a, b;
    a.x = a1p[ka] * sA;
    a.y = a1p[ka + 1] * sA;
    b.x = B1[(long long)ka * N + col];
    b.y = B1[(long long)(ka + 1) * N + col];
    c = __builtin_amdgcn_wmma_f32_16x16x4_f32(false, a, false, b,
                                              (short)0, c, false, false);
  }
  // pass 2: A2 @ B2
#pragma unroll 4
  for (int k = 0; k < K; k += 4) {
    int ka = k + (half << 1);
    v2f a, b;
    a.x = a2p[ka];
    a.y = a2p[ka + 1];
    b.x = B2[(long long)ka * N + col];
    b.y = B2[(long long)(ka + 1) * N + col];
    c = __builtin_amdgcn_wmma_f32_16x16x4_f32(false, a, false, b,
                                              (short)0, c, false, false);
  }

  // store: VGPR r -> row tm + r + 8*half
#pragma unroll
  for (int r = 0; r < 8; ++r) {
    int row = tm + r + (half << 3);
    C[(long long)row * N + col] = c[r];
  }
}

// ---------------- BatchNorm (train) + ReLU ----------------
__global__ void col_stats_kernel(const float* __restrict__ h,
                                 float* __restrict__ mean,
                                 float* __restrict__ var,
                                 int M, int N) {
  __shared__ float ss[256];
  __shared__ float ss2[256];
  int col = blockIdx.x;
  float s = 0.f, s2 = 0.f;
  for (int r = threadIdx.x; r < M; r += blockDim.x) {
    float v = h[(long long)r * N + col];
    s += v;
    s2 += v * v;
  }
  ss[threadIdx.x] = s;
  ss2[threadIdx.x] = s2;
  __syncthreads();
  for (int off = blockDim.x >> 1; off > 0; off >>= 1) {
    if ((int)threadIdx.x < off) {
      ss[threadIdx.x] += ss[threadIdx.x + off];
      ss2[threadIdx.x] += ss2[threadIdx.x + off];
    }
    __syncthreads();
  }
  if (threadIdx.x == 0) {
    float m = ss[0] / (float)M;
    mean[col] = m;
    var[col] = ss2[0] / (float)M - m * m;
  }
}

__global__ void bn_relu_kernel(float* __restrict__ h,
                               const float* __restrict__ gamma,
                               const float* __restrict__ beta,
                               const float* __restrict__ mean,
                               const float* __restrict__ var,
                               long long total, int N) {
  long long i = (long long)blockIdx.x * blockDim.x + threadIdx.x;
  if (i >= total) return;
  int col = (int)(i % N);
  float v = (h[i] - mean[col]) * rsqrtf(var[col] + EPS) * gamma[col] + beta[col];
  h[i] = v > 0.f ? v : 0.f;
}

// ---------------- host orchestration ----------------
static inline int ceil_div_ll(long long a, int b) { return (int)((a + b - 1) / b); }

extern "C" void kernel_launch(void* const* d_in, const int* in_sizes, int n_in,
                              void* d_out, int out_size, void* d_ws, size_t ws_size,
                              hipStream_t stream) {
  (void)in_sizes; (void)n_in; (void)out_size; (void)ws_size;

  const float* x    = (const float*)d_in[0];
  const float* Wl0  = (const float*)d_in[1];
  const float* bl0  = (const float*)d_in[2];
  const float* Wr0  = (const float*)d_in[3];
  const float* Wl1  = (const float*)d_in[4];
  const float* bl1  = (const float*)d_in[5];
  const float* Wr1  = (const float*)d_in[6];
  const float* Wl2  = (const float*)d_in[7];
  const float* bl2  = (const float*)d_in[8];
  const float* Wr2  = (const float*)d_in[9];
  const float* g0   = (const float*)d_in[10];
  const float* b0   = (const float*)d_in[11];
  const float* g1   = (const float*)d_in[12];
  const float* b1   = (const float*)d_in[13];
  const int* src0   = (const int*)d_in[14];
  const int* dst0   = (const int*)d_in[15];
  const int* src1   = (const int*)d_in[16];
  const int* dst1   = (const int*)d_in[17];
  const int* src2   = (const int*)d_in[18];
  const int* dst2   = (const int*)d_in[19];

  const int ND0 = 25600, ND1 = 5120, ND2 = 1024;
  const int E0 = 256000, E1 = 51200, E2 = 10240;
  const int IN = 128, H = 256, OUT = 64;

  // workspace carve-up (floats)
  float* w = (float*)d_ws;
  float* agg0 = w;  w += (long long)ND0 * IN;
  float* cnt0 = w;  w += ND0;
  float* h0   = w;  w += (long long)ND0 * H;
  float* agg1 = w;  w += (long long)ND1 * H;
  float* cnt1 = w;  w += ND1;
  float* h1   = w;  w += (long long)ND1 * H;
  float* agg2 = w;  w += (long long)ND2 * H;
  float* cnt2 = w;  w += ND2;
  float* mean = w;  w += H;
  float* var  = w;  w += H;

  // ---------------- layer 0 ----------------
  hipMemsetAsync(agg0, 0, sizeof(float) * ((size_t)ND0 * IN + ND0), stream);
  {
    long long tot = (long long)E0 * IN;
    scatter_add_kernel<<<ceil_div_ll(tot, 256), 256, 0, stream>>>(x, src0, dst0, agg0, tot, IN);
    count_kernel<<<(E0 + 255) / 256, 256, 0, stream>>>(dst0, cnt0, E0);
    inv_count_kernel<<<(ND0 + 255) / 256, 256, 0, stream>>>(cnt0, ND0);
    int tiles = (ND0 / 16) * (H / 16);                 // 25600 waves
    gemm_dual_wmma<<<(tiles + 7) / 8, 256, 0, stream>>>(agg0, cnt0, x, Wl0, Wr0, bl0,
                                                        h0, ND0, H, IN);
    col_stats_kernel<<<H, 256, 0, stream>>>(h0, mean, var, ND0, H);
    long long tot2 = (long long)ND0 * H;
    bn_relu_kernel<<<ceil_div_ll(tot2, 256), 256, 0, stream>>>(h0, g0, b0, mean, var, tot2, H);
  }

  // ---------------- layer 1 ----------------
  hipMemsetAsync(agg1, 0, sizeof(float) * ((size_t)ND1 * H + ND1), stream);
  {
    long long tot = (long long)E1 * H;
    scatter_add_kernel<<<ceil_div_ll(tot, 256), 256, 0, stream>>>(h0, src1, dst1, agg1, tot, H);
    count_kernel<<<(E1 + 255) / 256, 256, 0, stream>>>(dst1, cnt1, E1);
    inv_count_kernel<<<(ND1 + 255) / 256, 256, 0, stream>>>(cnt1, ND1);
    int tiles = (ND1 / 16) * (H / 16);                 // 5120 waves
    gemm_dual_wmma<<<(tiles + 7) / 8, 256, 0, stream>>>(agg1, cnt1, h0, Wl1, Wr1, bl1,
                                                        h1, ND1, H, H);
    col_stats_kernel<<<H, 256, 0, stream>>>(h1, mean, var, ND1, H);
    long long tot2 = (long long)ND1 * H;
    bn_relu_kernel<<<ceil_div_ll(tot2, 256), 256, 0, stream>>>(h1, g1, b1, mean, var, tot2, H);
  }

  // ---------------- layer 2 (no BN/ReLU, writes d_out) ----------------
  hipMemsetAsync(agg2, 0, sizeof(float) * ((size_t)ND2 * H + ND2), stream);
  {
    long long tot = (long long)E2 * H;
    scatter_add_kernel<<<ceil_div_ll(tot, 256), 256, 0, stream>>>(h1, src2, dst2, agg2, tot, H);
    count_kernel<<<(E2 + 255) / 256, 256, 0, stream>>>(dst2, cnt2, E2);
    inv_count_kernel<<<(ND2 + 255) / 256, 256, 0, stream>>>(cnt2, ND2);
    int tiles = (ND2 / 16) * (OUT / 16);               // 256 waves
    gemm_dual_wmma<<<(tiles + 7) / 8, 256, 0, stream>>>(agg2, cnt2, h1, Wl2, Wr2, bl2,
                                                        (float*)d_out, ND2, OUT, H);
  }
}